// DIN_80719615361480
// MI455X (gfx1250) — compile-verified
//
#include <hip/hip_runtime.h>
#include <hip/hip_bf16.h>
#include <math.h>

typedef __attribute__((ext_vector_type(2))) float v2f;
typedef __attribute__((ext_vector_type(8))) float v8f;

#define B_   1024
#define N_   200
#define K_   32
#define HID_ 36
#define NPAD 208   // 13 tiles of 16 rows
#define HPAD 48    // 3 tiles of 16 cols

__global__ __launch_bounds__(256) void din_factored_wmma(
    const float* __restrict__ cand,   // [B,32]  attention key k
    const float* __restrict__ goods,  // [B,200,32]
    const float* __restrict__ W1,     // [1088,36] rows: 0..31=k, 32..63=q, 64+i*32+j=outer
    const float* __restrict__ b1,     // [36]
    const float* __restrict__ W2,     // [36]
    const float* __restrict__ b2,     // [1]
    float* __restrict__ out)          // [B,32]
{
    __shared__ float kS[32];
    __shared__ float biasS[HPAD];
    __shared__ float w2S[HPAD];
    __shared__ float WeffS[32 * HPAD];
    __shared__ float goodsS[NPAD * 32];
    __shared__ float weightS[NPAD];
    __shared__ float outS[32];

    const int b      = blockIdx.x;
    const int tid    = threadIdx.x;
    const int lane   = tid & 31;
    const int lrow   = lane & 15;   // position within 16-lane half
    const int laneHi = lane >> 4;   // which half of the wave
    // wave index as a provably wave-uniform (SGPR) value: keeps the WMMA tile
    // loop scalar-uniform -> no EXEC masking around v_wmma (EXEC must be all 1s).
    const int wave   = __builtin_amdgcn_readfirstlane(tid >> 5);

    if (tid < 32) { kS[tid] = cand[b * 32 + tid]; outS[tid] = 0.f; }
    if (tid < NPAD) weightS[tid] = 0.f;
    __syncthreads();

    // bias_eff[h] = b1[h] + k . W1_k[:,h]   ;  W2 zero-padded to 48
    if (tid < HPAD) {
        int h = tid;
        float v = 0.f, wv = 0.f;
        if (h < HID_) {
            v = b1[h];
            #pragma unroll 8
            for (int i = 0; i < 32; ++i) v += kS[i] * W1[i * HID_ + h];
            wv = W2[h];
        }
        biasS[h] = v;
        w2S[h]   = wv;
    }

    // Weff[j][h] = W1_q[j][h] + sum_i k[i] * W1_outer[i*32+j][h]  (cols >=36 -> 0)
    for (int e = tid; e < 32 * HPAD; e += 256) {
        int j = e / HPAD, h = e - j * HPAD;
        float v = 0.f;
        if (h < HID_) {
            v = W1[(32 + j) * HID_ + h];
            #pragma unroll 8
            for (int i = 0; i < 32; ++i)
                v += kS[i] * W1[(64 + i * 32 + j) * HID_ + h];
        }
        WeffS[j * HPAD + h] = v;
    }

    // Stage goods[b] -> LDS (rows 200..207 zeroed), 128-bit loads
    {
        const float4* g4 = (const float4*)(goods + (size_t)b * N_ * K_);
        float4* s4 = (float4*)goodsS;
        for (int e = tid; e < NPAD * 32 / 4; e += 256) {
            int row = e >> 3;                    // 8 float4 per 32-wide row
            float4 v = make_float4(0.f, 0.f, 0.f, 0.f);
            if (row < N_) v = g4[e];
            s4[e] = v;
        }
    }
    __syncthreads();

    // GEMM: act_pre[208x48] = goods[208x32] @ Weff[32x48], via fp32 WMMA 16x16x4.
    // Fused epilogue: sigmoid(act_pre + bias) * W2pad, reduced over h into weight[n].
    for (int tile = wave; tile < 13 * 3; tile += 8) {
        int mt = tile / 3, nt = tile - mt * 3;
        int mBase = mt * 16, nBase = nt * 16;
        v8f acc = {};
        #pragma unroll
        for (int kk = 0; kk < 32; kk += 4) {
            // A 16x4 layout: lanes 0-15 hold K=kk,kk+1; lanes 16-31 hold K=kk+2,kk+3
            int ka = kk + laneHi * 2;
            v2f a, bf;
            a.x  = goodsS[(mBase + lrow) * 32 + ka];
            a.y  = goodsS[(mBase + lrow) * 32 + ka + 1];
            // B 4x16 layout: v0 = rows kk (lanes 0-15) / kk+2 (lanes 16-31); v1 = +1
            bf.x = WeffS[ ka      * HPAD + nBase + lrow];
            bf.y = WeffS[(ka + 1) * HPAD + nBase + lrow];
            acc = __builtin_amdgcn_wmma_f32_16x16x4_f32(
                      false, a, false, bf, (short)0, acc, false, false);
        }
        // C/D layout: vgpr r, lanes 0-15 -> m=mBase+r, lanes 16-31 -> m=mBase+r+8; n=nBase+lrow
        int   h    = nBase + lrow;
        float bias = biasS[h];
        float w2v  = w2S[h];
        #pragma unroll
        for (int r = 0; r < 8; ++r) {
            float x = acc[r] + bias;
            // sigmoid via v_exp_f32 + v_rcp_f32 (avoid IEEE div expansion)
            float s = __builtin_amdgcn_rcpf(1.0f + __expf(-x));
            float c = s * w2v;
            c += __shfl_xor(c, 1, 32);
            c += __shfl_xor(c, 2, 32);
            c += __shfl_xor(c, 4, 32);
            c += __shfl_xor(c, 8, 32);   // full sum within each 16-lane half
            if (lrow == 0)
                atomicAdd(&weightS[mBase + r + 8 * laneHi], c);
        }
    }
    __syncthreads();

    float b2v = b2[0];
    if (tid < NPAD) weightS[tid] += b2v;
    __syncthreads();

    // out[b][j] = sum_{n<200} goods[n][j] * weight[n]
    float partial = 0.f;
    for (int n = wave; n < N_; n += 8)
        partial += goodsS[n * 32 + lane] * weightS[n];
    atomicAdd(&outS[lane], partial);
    __syncthreads();
    if (tid < 32) out[b * 32 + tid] = outS[tid];
}

extern "C" void kernel_launch(void* const* d_in, const int* in_sizes, int n_in,
                              void* d_out, int out_size, void* d_ws, size_t ws_size,
                              hipStream_t stream) {
    const float* cand  = (const float*)d_in[0];
    const float* goods = (const float*)d_in[1];
    const float* W1    = (const float*)d_in[2];
    const float* b1    = (const float*)d_in[3];
    const float* W2    = (const float*)d_in[4];
    const float* b2    = (const float*)d_in[5];
    float* out = (float*)d_out;

    din_factored_wmma<<<dim3(B_), dim3(256), 0, stream>>>(
        cand, goods, W1, b1, W2, b2, out);
}